// MatrixNet_29815662969326
// MI455X (gfx1250) — compile-verified
//
#include <hip/hip_runtime.h>
#include <math.h>

// fp32 WMMA fragment types: A/B = 16x4 / 4x16 f32 -> 2 VGPRs/lane; C/D 16x16 f32 -> 8 VGPRs/lane
typedef __attribute__((ext_vector_type(2))) float v2f;
typedef __attribute__((ext_vector_type(8))) float v8f;

#define NGRID   10
#define NCELLS  100     // unique patches
#define MPAD    112     // 7 M-tiles of 16
#define KPATCH  28      // 27 padded to 28 (multiple of 4)
#define N_H0    32
#define N_H1    256
#define N_H2    64

struct Smem {
  float padded[12 * 12 * 3];        // zero-padded HWC image
  float P[MPAD * KPATCH];           // patch matrix (A of GEMM1)
  float WcB[KPATCH * N_H0];         // Wc^T zero-padded to 28x32 (B of GEMM1)
  float H0[MPAD * N_H0];            // 32-wide activations
  float H2[MPAD * N_H2];            // 64-wide activations
  float E[MPAD];                    // final embeddings (scalar per patch)
};                                  // 15,328 floats = 61,312 B LDS

// One 16x16 output tile, K-loop of fp32 WMMAs. A is row-major (lda), generic ptr
// (LDS via flat or global both fine). bload(k, nLocal) supplies B[k][nBase+nLocal].
template <typename BLoad>
__device__ __forceinline__ v8f wmma_tile(const float* __restrict__ A, int lda, int K,
                                         int mBase, BLoad bload, v8f acc) {
  const int lane  = threadIdx.x & 31;
  const int kOff  = (lane >> 4) << 1;   // A/B frag: lanes 0-15 -> K {0,1}; 16-31 -> K {2,3}
  const int lr    = lane & 15;
  const float* __restrict__ arow = A + (size_t)(mBase + lr) * lda;
  for (int k0 = 0; k0 < K; k0 += 4) {
    v2f a, b;
    a.x = arow[k0 + kOff];
    a.y = arow[k0 + kOff + 1];
    b.x = bload(k0 + kOff,     lr);
    b.y = bload(k0 + kOff + 1, lr);
    // v_wmma_f32_16x16x4_f32: (neg_a, A, neg_b, B, c_mod, C, reuse_a, reuse_b)
    acc = __builtin_amdgcn_wmma_f32_16x16x4_f32(false, a, false, b, (short)0, acc,
                                                false, false);
  }
  return acc;
}

// C/D layout (ISA 7.12.2): VGPR r -> row r (lanes 0-15) / row r+8 (lanes 16-31), col = lane%16
__device__ __forceinline__ void store_tile(float* __restrict__ D, int ldd, int mBase,
                                           int nBase, v8f acc, bool relu) {
  const int lane = threadIdx.x & 31;
  const int half = lane >> 4;
  const int lr   = lane & 15;
#pragma unroll
  for (int r = 0; r < 8; ++r) {
    float v = acc[r];
    if (relu) v = fmaxf(v, 0.0f);
    D[(size_t)(mBase + r + half * 8) * ldd + (nBase + lr)] = v;
  }
}

__global__ void __launch_bounds__(256, 1)
matrixnet_fused(const float* __restrict__ image,  // (3,10,10)
                const float* __restrict__ Wc,     // (32,3,3,3) = (32,27)
                const float* __restrict__ bc,     // (32,)
                const float* __restrict__ W1,     // (256,32)
                const float* __restrict__ b1,     // (256,)
                const float* __restrict__ W2,     // (64,256)
                const float* __restrict__ b2,     // (64,)
                const float* __restrict__ W3,     // (1,64)
                const float* __restrict__ b3,     // (1,)
                const int*   __restrict__ pair_i,
                const int*   __restrict__ pair_j,
                float* __restrict__ out,          // (100,100)
                float* __restrict__ ws,           // H1: 112x256 f32 in global scratch
                int npairs) {
  __shared__ Smem s;
  const int tid  = threadIdx.x;
  const int lane = tid & 31;
  const int lr   = lane & 15;
  const int wave = tid >> 5;            // 8 wave32s

  // Warm L2/L0 for the two big weight matrices (gfx1250 global_prefetch_b8 path).
  for (int t = tid; t < (int)((N_H1 * N_H0 + N_H2 * N_H1) * sizeof(float)) / 128;
       t += blockDim.x) {
    const char* base = (t < (N_H1 * N_H0 * 4) / 128)
                           ? (const char*)W1 + t * 128
                           : (const char*)W2 + (t - (N_H1 * N_H0 * 4) / 128) * 128;
    __builtin_prefetch(base, 0, 3);
  }

  // Zero the output (reference output is zeros except scattered pairs).
  for (int t = tid; t < NCELLS * NCELLS; t += blockDim.x) out[t] = 0.0f;

  // ---- Stage 0: padded HWC image, Wc^T panel, patch matrix P (112x28, zero padded) ----
  for (int t = tid; t < 12 * 12 * 3; t += blockDim.x) s.padded[t] = 0.0f;
  for (int t = tid; t < KPATCH * N_H0; t += blockDim.x) {
    int k = t / N_H0, n = t % N_H0;               // B panel [k][n], zero rows 27
    s.WcB[t] = (k < 27) ? Wc[n * 27 + k] : 0.0f;
  }
  __syncthreads();
  for (int t = tid; t < 3 * NGRID * NGRID; t += blockDim.x) {
    int c = t / (NGRID * NGRID), rem = t % (NGRID * NGRID);
    int y = rem / NGRID, x = rem % NGRID;
    s.padded[((y + 1) * 12 + (x + 1)) * 3 + c] = image[t];
  }
  __syncthreads();
  for (int e = tid; e < MPAD * KPATCH; e += blockDim.x) {
    int m = e / KPATCH, k = e % KPATCH;
    float v = 0.0f;
    if (m < NCELLS && k < 27) {
      int a = k / 9, b = (k % 9) / 3, c = k % 3;
      int xq = m / NGRID, yq = m % NGRID;        // reference: rows<-y, cols<-x
      v = s.padded[((yq + a) * 12 + (xq + b)) * 3 + c];
    }
    s.P[e] = v;
  }
  __syncthreads();

  // ---- GEMM1: H0 = relu(P(112x28) x WcB(28x32) + bc) ---- 14 tiles, branch-free B in LDS
  for (int t = wave; t < 7 * 2; t += 8) {
    int mB = (t % 7) * 16, nB = (t / 7) * 16;
    float bv = bc[nB + lr];
    v8f acc = {bv, bv, bv, bv, bv, bv, bv, bv};
    acc = wmma_tile(s.P, KPATCH, KPATCH, mB,
                    [&](int k, int n) { return s.WcB[k * N_H0 + nB + n]; }, acc);
    store_tile(s.H0, N_H0, mB, nB, acc, true);
  }
  __syncthreads();

  // ---- GEMM2: H1 = relu(H0(112x32) x W1^T(32x256) + b1) ---- 112 tiles -> global ws
  for (int t = wave; t < 7 * 16; t += 8) {
    int mB = (t % 7) * 16, nB = (t / 7) * 16;
    float bv = b1[nB + lr];
    v8f acc = {bv, bv, bv, bv, bv, bv, bv, bv};
    acc = wmma_tile(s.H0, N_H0, N_H0, mB,
                    [&](int k, int n) { return W1[(nB + n) * N_H0 + k]; }, acc);
    store_tile(ws, N_H1, mB, nB, acc, true);
  }
  __threadfence();
  __syncthreads();

  // ---- GEMM3: H2 = relu(H1(112x256) x W2^T(256x64) + b2) ---- 28 tiles, K=256
  for (int t = wave; t < 7 * 4; t += 8) {
    int mB = (t % 7) * 16, nB = (t / 7) * 16;
    float bv = b2[nB + lr];
    v8f acc = {bv, bv, bv, bv, bv, bv, bv, bv};
    acc = wmma_tile(ws, N_H1, N_H1, mB,
                    [&](int k, int n) { return W2[(nB + n) * N_H1 + k]; }, acc);
    store_tile(s.H2, N_H2, mB, nB, acc, true);
  }
  __syncthreads();

  // ---- GEMM4 (matvec, N=1): E = H2(100x64) x W3^T + b3 ----
  if (tid < NCELLS) {
    float e = b3[0];
#pragma unroll 8
    for (int k = 0; k < N_H2; ++k) e += s.H2[tid * N_H2 + k] * W3[k];
    s.E[tid] = e;
  }
  __syncthreads();

  // ---- Scatter: out[i,j] = sigmoid(E[i] * E[(i/10)*10 + j%10]) ----
  for (int t = tid; t < npairs; t += blockDim.x) {
    int i = pair_i[t], j = pair_j[t];
    int k = (i / NGRID) * NGRID + (j % NGRID);
    float z = s.E[i] * s.E[k];
    out[i * NCELLS + j] = 1.0f / (1.0f + expf(-z));
  }
}

extern "C" void kernel_launch(void* const* d_in, const int* in_sizes, int n_in,
                              void* d_out, int out_size, void* d_ws, size_t ws_size,
                              hipStream_t stream) {
  const float* image = (const float*)d_in[0];
  const float* Wc    = (const float*)d_in[1];
  const float* bc    = (const float*)d_in[2];
  const float* W1    = (const float*)d_in[3];
  const float* b1    = (const float*)d_in[4];
  const float* W2    = (const float*)d_in[5];
  const float* b2    = (const float*)d_in[6];
  const float* W3    = (const float*)d_in[7];
  const float* b3    = (const float*)d_in[8];
  const int*   pi    = (const int*)d_in[9];
  const int*   pj    = (const int*)d_in[10];
  int npairs = in_sizes[9];

  // Single workgroup (8 wave32s) — ~5 MFLOP workload is pure latency; one launch, no grid.
  matrixnet_fused<<<1, 256, 0, stream>>>(image, Wc, bc, W1, b1, W2, b2, W3, b3,
                                         pi, pj, (float*)d_out, (float*)d_ws, npairs);
}